// OODDetector_61108794687905
// MI455X (gfx1250) — compile-verified
//
#include <hip/hip_runtime.h>

// GCN autoencoder for MI455X (gfx1250, wave32).
// Dense transforms via V_WMMA_F32_16X16X4_F32 (fp32 precision, GEMMs are
// compute-trivial here); edge aggregation via coalesced float4 gathers +
// global f32 atomics (L2-resident working set, 192MB L2 >> 38MB features).

typedef __attribute__((ext_vector_type(2))) float v2f;
typedef __attribute__((ext_vector_type(8))) float v8f;

// ---------------------------------------------------------------------------
// Generic fp32 WMMA GEMM: C[nrows x DOUT] = op_A(A[nrows x K]) @ B[K x DOUT]
// op_A = identity, or relu(A + biasA[k]) fused into the operand fetch.
// Optional per-column biasC added at store.  One 16x16 tile per wave,
// 8 waves / 256 threads per block, B staged in LDS.
// ---------------------------------------------------------------------------
template<int K, int DOUT, bool BIASRELU_A, bool BIAS_C>
__global__ __launch_bounds__(256)
void gemm_wmma_f32(const float* __restrict__ A,
                   const float* __restrict__ B,
                   const float* __restrict__ biasA,
                   const float* __restrict__ biasC,
                   float* __restrict__ C,
                   int nrows)
{
    constexpr int TILES_C = DOUT / 16;       // col tiles per row tile
    constexpr int TILES_R = 8 / TILES_C;     // row tiles per block (8 waves)

    __shared__ float Bs[K * DOUT];
    for (int i = threadIdx.x; i < K * DOUT; i += 256) Bs[i] = B[i];
    __syncthreads();

    const int wave = threadIdx.x >> 5;       // wave32
    const int lane = threadIdx.x & 31;
    const int half = lane >> 4;              // selects K-pair (A) / K-row pair (B)
    const int l16  = lane & 15;

    const int row_tile = blockIdx.x * TILES_R + (wave / TILES_C);
    const int col0     = (wave % TILES_C) * 16;
    const int r0       = row_tile * 16;
    if (r0 >= nrows) return;                 // wave-uniform: EXEC stays all-1s

    const float* Arow = A + (size_t)(r0 + l16) * K;

    v8f acc = {};
#pragma unroll
    for (int k0 = 0; k0 < K; k0 += 4) {
        const int ka = k0 + half * 2;
        float a0 = Arow[ka];
        float a1 = Arow[ka + 1];
        if (BIASRELU_A) {
            a0 = fmaxf(a0 + biasA[ka], 0.0f);
            a1 = fmaxf(a1 + biasA[ka + 1], 0.0f);
        }
        v2f a, b;
        a[0] = a0;
        a[1] = a1;
        b[0] = Bs[ka * DOUT + col0 + l16];
        b[1] = Bs[(ka + 1) * DOUT + col0 + l16];
        acc = __builtin_amdgcn_wmma_f32_16x16x4_f32(
            /*neg_a=*/false, a, /*neg_b=*/false, b,
            /*c_mod=*/(short)0, acc, /*reuse_a=*/false, /*reuse_b=*/false);
    }

    const float cb = BIAS_C ? biasC[col0 + l16] : 0.0f;
#pragma unroll
    for (int i = 0; i < 8; ++i) {
        const int row = r0 + i + half * 8;   // C/D layout: VGPR i -> M=i / M=i+8
        C[(size_t)row * DOUT + col0 + l16] = acc[i] + cb;
    }
}

// ---------------------------------------------------------------------------
// Degree / normalization helpers
// ---------------------------------------------------------------------------
__global__ void k_fill1(float* __restrict__ p, int n)
{
    int i = blockIdx.x * blockDim.x + threadIdx.x;
    if (i < n) p[i] = 1.0f;                  // self-loop contributes 1 to degree
}

__global__ void k_deg_count(const long long* __restrict__ dst,
                            float* __restrict__ deg, int E)
{
    int e = blockIdx.x * blockDim.x + threadIdx.x;
    if (e < E) atomicAdd(&deg[dst[e]], 1.0f);
}

__global__ void k_rsqrt_inplace(float* __restrict__ p, int n)
{
    int i = blockIdx.x * blockDim.x + threadIdx.x;
    if (i < n) p[i] = rsqrtf(p[i]);          // deg >= 1 always (self-loop)
}

// out[i][:] = h[i][:] * dinv[i]^2   (self-loop message; also zero-inits accum)
template<int D>
__global__ void k_selfloop_init(const float* __restrict__ h,
                                const float* __restrict__ dinv,
                                float* __restrict__ out, int N)
{
    int idx = blockIdx.x * blockDim.x + threadIdx.x;
    if (idx >= N * D) return;
    const int i = idx / D;
    const float di = dinv[i];
    out[idx] = h[idx] * di * di;
}

// Edge scatter: out[dst] += h[src] * dinv[src]*dinv[dst].
// D/4 threads per edge, float4 gathers (coalesced), 4 f32 atomics per thread.
template<int D>
__global__ void k_edge_scatter(const long long* __restrict__ src,
                               const long long* __restrict__ dst,
                               const float* __restrict__ dinv,
                               const float* __restrict__ h,
                               float* __restrict__ out, int E)
{
    constexpr int TPE = D / 4;
    int idx = blockIdx.x * blockDim.x + threadIdx.x;
    int e = idx / TPE;
    int t = idx % TPE;
    if (e >= E) return;
    const long long s = src[e];
    const long long d = dst[e];
    const float norm = dinv[s] * dinv[d];
    const float4 hv = reinterpret_cast<const float4*>(h + (size_t)s * D)[t];
    float* o = out + (size_t)d * D + t * 4;
    atomicAdd(o + 0, hv.x * norm);
    atomicAdd(o + 1, hv.y * norm);
    atomicAdd(o + 2, hv.z * norm);
    atomicAdd(o + 3, hv.w * norm);
}

// z[i][j] += b[j]  (conv2 bias, in place; no relu on the latent)
template<int D>
__global__ void k_bias_add(float* __restrict__ z, const float* __restrict__ b, int n)
{
    int idx = blockIdx.x * blockDim.x + threadIdx.x;
    if (idx < n) z[idx] += b[idx % D];
}

// ---------------------------------------------------------------------------
static inline int cdiv(long long a, long long b) { return (int)((a + b - 1) / b); }

extern "C" void kernel_launch(void* const* d_in, const int* in_sizes, int n_in,
                              void* d_out, int out_size, void* d_ws, size_t ws_size,
                              hipStream_t stream)
{
    (void)n_in; (void)out_size; (void)ws_size;

    const float*     x   = (const float*)d_in[0];
    const long long* ei  = (const long long*)d_in[1];   // int64 [2, E]
    const float*     W1  = (const float*)d_in[2];       // [128,64]
    const float*     b1  = (const float*)d_in[3];       // [64]
    const float*     W2  = (const float*)d_in[4];       // [64,32]
    const float*     b2  = (const float*)d_in[5];       // [32]
    const float*     Wd1 = (const float*)d_in[6];       // [32,64]
    const float*     bd1 = (const float*)d_in[7];       // [64]
    const float*     Wd2 = (const float*)d_in[8];       // [64,128]
    const float*     bd2 = (const float*)d_in[9];       // [128]

    const int N = in_sizes[0] / 128;
    const int E = in_sizes[1] / 2;
    const long long* src = ei;
    const long long* dst = ei + E;

    float* recon = (float*)d_out;                 // [N,128]
    float* z     = recon + (size_t)N * 128;       // [N,32]

    // Workspace layout (fp32): dinv[N] | h1[N,64] (reused as decoder hidden)
    //                          | agg1[N,64] | h2[N,32]
    float* dinv = (float*)d_ws;
    float* h1   = dinv + (((size_t)N + 255) & ~(size_t)255);
    float* agg1 = h1 + (size_t)N * 64;
    float* h2   = agg1 + (size_t)N * 64;

    // ---- symmetric normalization: dinv = rsqrt(indeg + 1) ----
    k_fill1<<<cdiv(N, 256), 256, 0, stream>>>(dinv, N);
    k_deg_count<<<cdiv(E, 256), 256, 0, stream>>>(dst, dinv, E);
    k_rsqrt_inplace<<<cdiv(N, 256), 256, 0, stream>>>(dinv, N);

    // ---- GCNConv 1: h1 = x @ W1 ; agg1 = scatter(norm * h1) ----
    gemm_wmma_f32<128, 64, false, false>
        <<<cdiv(N, 32), 256, 0, stream>>>(x, W1, nullptr, nullptr, h1, N);
    k_selfloop_init<64>
        <<<cdiv((long long)N * 64, 256), 256, 0, stream>>>(h1, dinv, agg1, N);
    k_edge_scatter<64>
        <<<cdiv((long long)E * 16, 256), 256, 0, stream>>>(src, dst, dinv, h1, agg1, E);

    // ---- GCNConv 2: h2 = relu(agg1 + b1) @ W2  (bias+relu fused in A fetch) ----
    gemm_wmma_f32<64, 32, true, false>
        <<<cdiv(N, 64), 256, 0, stream>>>(agg1, W2, b1, nullptr, h2, N);
    k_selfloop_init<32>
        <<<cdiv((long long)N * 32, 256), 256, 0, stream>>>(h2, dinv, z, N);
    k_edge_scatter<32>
        <<<cdiv((long long)E * 8, 256), 256, 0, stream>>>(src, dst, dinv, h2, z, E);
    k_bias_add<32>
        <<<cdiv((long long)N * 32, 256), 256, 0, stream>>>(z, b2, N * 32);

    // ---- Decoder: h = z @ Wd1 (raw, reuse h1) ;
    //      recon = relu(h + bd1) @ Wd2 + bd2 (both biases fused) ----
    gemm_wmma_f32<32, 64, false, false>
        <<<cdiv(N, 32), 256, 0, stream>>>(z, Wd1, nullptr, nullptr, h1, N);
    gemm_wmma_f32<64, 128, true, true>
        <<<cdiv(N, 16), 256, 0, stream>>>(h1, Wd2, bd1, bd2, recon, N);
}